// BDLoss_29712583753730
// MI455X (gfx1250) — compile-verified
//
#include <hip/hip_runtime.h>
#include <cstdint>
#include <cstddef>

// Problem constants (match the reference)
#define BB   2
#define CCH  4
#define XX   128
#define YY   128
#define ZZ   96
#define VOL  (XX * YY * ZZ)          // 1,572,864
#define BIGF 1e10f
#define NB_ACC 256                   // accumulation blocks per group
#define NGRP   6                     // (b,c) groups: 2 batches x classes 1..3
#define MEAN_DEN 9437184.0f          // B*(C-1)*VOL = 2*3*1572864

typedef float v2f __attribute__((ext_vector_type(2)));
typedef float v8f __attribute__((ext_vector_type(8)));

#if __has_builtin(__builtin_amdgcn_wmma_f32_16x16x4_f32)
#define HAVE_WMMA4 1
#else
#define HAVE_WMMA4 0
#endif

#if __has_builtin(__builtin_amdgcn_global_load_async_to_lds_b32)
#define HAVE_ASYNC 1
#else
#define HAVE_ASYNC 0
#endif

typedef __attribute__((address_space(1))) int* gas_i32p;
typedef __attribute__((address_space(3))) int* las_i32p;

static __device__ __forceinline__ float readlane_f(float v, int lane) {
  return __builtin_bit_cast(
      float, __builtin_amdgcn_readlane(__builtin_bit_cast(unsigned int, v),
                                       (unsigned int)lane));
}

// ---------------------------------------------------------------------------
// 0) zero the per-(b,c) "class present" flags
// ---------------------------------------------------------------------------
__global__ void zero_flags_kernel(int* flags) {
  if (threadIdx.x < BB * CCH) flags[threadIdx.x] = 0;
}

// ---------------------------------------------------------------------------
// 1) any_pos flags: does class c appear anywhere in gt[b]?
// ---------------------------------------------------------------------------
__global__ __launch_bounds__(256) void flags_kernel(const int* __restrict__ gt,
                                                    int* __restrict__ flags) {
  __shared__ int lf[BB * CCH];
  if (threadIdx.x < BB * CCH) lf[threadIdx.x] = 0;
  __syncthreads();
  const int total = BB * VOL;
  for (int idx = blockIdx.x * blockDim.x + threadIdx.x; idx < total;
       idx += gridDim.x * blockDim.x) {
    int b = idx / VOL;
    int g = gt[idx];
    if (g > 0 && g < CCH) lf[b * CCH + g] = 1;  // benign same-value race
  }
  __syncthreads();
  if (threadIdx.x < BB * CCH && lf[threadIdx.x])
    atomicOr(&flags[threadIdx.x], 1);           // idempotent -> deterministic
}

// ---------------------------------------------------------------------------
// 2) init the two EDT source volumes for group (b,c):
//    pos = mask ? BIG : 0 ;  neg = mask ? 0 : BIG
// ---------------------------------------------------------------------------
__global__ __launch_bounds__(256) void init_kernel(const int* __restrict__ gt,
                                                   float* __restrict__ pos,
                                                   float* __restrict__ neg,
                                                   int b, int c) {
  const int* gtb = gt + (size_t)b * VOL;
  for (int p = blockIdx.x * blockDim.x + threadIdx.x; p < VOL;
       p += gridDim.x * blockDim.x) {
    bool m = (gtb[p] == c);
    pos[p] = m ? BIGF : 0.0f;
    neg[p] = m ? 0.0f : BIGF;
  }
}

// ---------------------------------------------------------------------------
// 3) one separable squared-EDT pass, in place.
//    Each block owns one whole line: stage it into LDS (async global->LDS on
//    CDNA5), then every thread i computes min_j (s[j] + (i-j)^2) with four
//    independent min chains (exact: (i-j)^2 <= 127^2 is integral in f32, so
//    fma == mul+add here, matching the reference arithmetic).
//    Grid covers both volumes: blockIdx.x = vol*linesPerVol + line.
// ---------------------------------------------------------------------------
__global__ __launch_bounds__(128) void edt_pass_kernel(float* __restrict__ vol0,
                                                       int L, int stride,
                                                       int linesPerVol,
                                                       int divi, int mulHi,
                                                       int mulLo) {
  __shared__ float s[128];
  const int line = blockIdx.x % linesPerVol;
  const int v    = blockIdx.x / linesPerVol;  // 0 = pos, 1 = neg
  float* f = vol0 + (size_t)v * VOL + (size_t)(line / divi) * mulHi +
             (size_t)(line % divi) * mulLo;
  const int i = threadIdx.x;

#if HAVE_ASYNC
  if (i < L) {
    float* gp = f + (size_t)i * stride;
    int* gpi = (int*)gp;              // bitcast in generic AS
    int* lpi = (int*)&s[i];
    gas_i32p g1 = (gas_i32p)gpi;      // addrspacecast -> global
    las_i32p l3 = (las_i32p)lpi;      // addrspacecast -> LDS
    __builtin_amdgcn_global_load_async_to_lds_b32(g1, l3, 0, 0);
  }
#if __has_builtin(__builtin_amdgcn_s_wait_asynccnt)
  __builtin_amdgcn_s_wait_asynccnt(0);
#else
  asm volatile("s_wait_asynccnt 0" ::: "memory");
#endif
  __syncthreads();
#else
  if (i < L) s[i] = f[(size_t)i * stride];
  __syncthreads();
#endif

  if (i < L) {
    float m0 = 3.0e38f, m1 = 3.0e38f, m2 = 3.0e38f, m3 = 3.0e38f;
    float d0 = (float)i, d1 = d0 - 1.0f, d2 = d0 - 2.0f, d3 = d0 - 3.0f;
    // L is 128 or 96: divisible by 4
    for (int j = 0; j < L; j += 4) {
      m0 = fminf(m0, fmaf(d0, d0, s[j + 0]));
      m1 = fminf(m1, fmaf(d1, d1, s[j + 1]));
      m2 = fminf(m2, fmaf(d2, d2, s[j + 2]));
      m3 = fminf(m3, fmaf(d3, d3, s[j + 3]));
      d0 -= 4.0f; d1 -= 4.0f; d2 -= 4.0f; d3 -= 4.0f;
    }
    f[(size_t)i * stride] = fminf(fminf(m0, m1), fminf(m2, m3));
  }
}

// ---------------------------------------------------------------------------
// 4) accumulation for group g=(b,c): softmax prob of channel c times
//    phi = present ? (boundary ? 0 : sqrt(neg)-sqrt(pos)) : 0.
//    Block partials are reduced per-wave with one V_WMMA_F32_16X16X4_F32
//    (A = lane partials, B = ones  => row sums), deterministically combined.
// ---------------------------------------------------------------------------
__global__ __launch_bounds__(256) void accum_kernel(
    const float* __restrict__ net, const int* __restrict__ gt,
    const float* __restrict__ pos, const float* __restrict__ neg,
    const int* __restrict__ flags, float* __restrict__ partials, int b, int c,
    int g) {
  float acc = 0.0f;
  const int present = flags[b * CCH + c];
  const int* gtb = gt + (size_t)b * VOL;
  const float* nb = net + (size_t)b * CCH * VOL;

  if (present) {
    for (int p = blockIdx.x * blockDim.x + threadIdx.x; p < VOL;
         p += gridDim.x * blockDim.x) {
      // softmax over 4 channels at voxel p
      float n0 = nb[p];
      float n1 = nb[p + VOL];
      float n2 = nb[p + 2 * VOL];
      float n3 = nb[p + 3 * VOL];
      float mx = fmaxf(fmaxf(n0, n1), fmaxf(n2, n3));
      float e0 = __expf(n0 - mx), e1 = __expf(n1 - mx);
      float e2 = __expf(n2 - mx), e3 = __expf(n3 - mx);
      float pc = (c == 1 ? e1 : (c == 2 ? e2 : e3)) / (e0 + e1 + e2 + e3);

      int gc = gtb[p];
      float phi;
      if (gc == c) {
        // inner boundary test (connectivity-1, edge replication => skip OOB)
        int z = p % ZZ;
        int y = (p / ZZ) % YY;
        int x = p / (YY * ZZ);
        bool bd = false;
        if (z > 0)      bd |= (gtb[p - 1] != c);
        if (z < ZZ - 1) bd |= (gtb[p + 1] != c);
        if (y > 0)      bd |= (gtb[p - ZZ] != c);
        if (y < YY - 1) bd |= (gtb[p + ZZ] != c);
        if (x > 0)      bd |= (gtb[p - YY * ZZ] != c);
        if (x < XX - 1) bd |= (gtb[p + YY * ZZ] != c);
        phi = bd ? 0.0f : (sqrtf(neg[p]) - sqrtf(pos[p]));
      } else {
        phi = sqrtf(neg[p]) - sqrtf(pos[p]);
      }
      acc += pc * phi;
    }
  }

  // ---- wave32 reduction via WMMA (EXEC is full here: all 256 threads live)
  __shared__ float wsum[8];
  const int lane = threadIdx.x & 31;
  const int wave = threadIdx.x >> 5;
  float w;
#if HAVE_WMMA4
  {
    v2f a;  a.x = acc;  a.y = 0.0f;    // A[M][k]: lane partial in K=0 / K=2
    v2f bmat; bmat.x = 1.0f; bmat.y = 1.0f;  // B = all ones (4x16)
    v8f cmat = {};
    cmat = __builtin_amdgcn_wmma_f32_16x16x4_f32(
        /*neg_a=*/false, a, /*neg_b=*/false, bmat,
        /*c_mod=*/(short)0, cmat, /*reuse_a=*/false, /*reuse_b=*/false);
    // lane j<16 column sum = sum_{M=0..7}(acc_M+acc_{M+16});
    // lane j>=16 column sum = sum_{M=8..15}(...); total = lane0 + lane16
    float t = cmat[0] + cmat[1] + cmat[2] + cmat[3] + cmat[4] + cmat[5] +
              cmat[6] + cmat[7];
    w = readlane_f(t, 0) + readlane_f(t, 16);
  }
#else
  {
    __shared__ float red[256];
    red[threadIdx.x] = acc;
    __syncthreads();
    for (int s = 128; s > 0; s >>= 1) {
      if (threadIdx.x < s) red[threadIdx.x] += red[threadIdx.x + s];
      __syncthreads();
    }
    w = red[0];  // fallback path only used if WMMA builtin missing
  }
#endif
  if (lane == 0) wsum[wave] = w;
  __syncthreads();
  if (threadIdx.x == 0) {
    float s = 0.0f;
#if HAVE_WMMA4
    for (int i = 0; i < 8; ++i) s += wsum[i];
#else
    s = wsum[0];
#endif
    partials[g * NB_ACC + blockIdx.x] = s;
  }
}

// ---------------------------------------------------------------------------
// 5) deterministic fixed-order final sum + mean scaling
// ---------------------------------------------------------------------------
__global__ void finish_kernel(const float* __restrict__ partials,
                              float* __restrict__ out, int n) {
  if (blockIdx.x == 0 && threadIdx.x == 0) {
    float s = 0.0f;
    for (int i = 0; i < n; ++i) s += partials[i];
    out[0] = s * (1.0f / MEAN_DEN);
  }
}

// ---------------------------------------------------------------------------
extern "C" void kernel_launch(void* const* d_in, const int* in_sizes, int n_in,
                              void* d_out, int out_size, void* d_ws,
                              size_t ws_size, hipStream_t stream) {
  (void)in_sizes; (void)n_in; (void)out_size; (void)ws_size;
  const float* net = (const float*)d_in[0];  // (B,C,X,Y,Z) f32
  const int*   gt  = (const int*)d_in[1];    // (B,1,X,Y,Z) i32
  float* out = (float*)d_out;

  // workspace layout (floats): [0..63] flags (ints), [64..1599] partials,
  // [4096 ..) pos volume, then neg volume. Total ~12.6 MB + 16 KB.
  int*   flags    = (int*)d_ws;
  float* partials = (float*)d_ws + 64;
  float* volPos   = (float*)d_ws + 4096;
  float* volNeg   = volPos + VOL;

  zero_flags_kernel<<<1, 32, 0, stream>>>(flags);
  flags_kernel<<<512, 256, 0, stream>>>(gt, flags);

  for (int g = 0; g < NGRP; ++g) {
    const int b = g / 3;
    const int c = 1 + (g % 3);
    init_kernel<<<1024, 256, 0, stream>>>(gt, volPos, volNeg, b, c);
    // X pass: L=128, stride=Y*Z, lines=Y*Z, base = line
    edt_pass_kernel<<<2 * YY * ZZ, 128, 0, stream>>>(volPos, XX, YY * ZZ,
                                                     YY * ZZ, 1, 1, 0);
    // Y pass: L=128, stride=Z, lines=X*Z, base = (l/Z)*(Y*Z) + (l%Z)
    edt_pass_kernel<<<2 * XX * ZZ, 128, 0, stream>>>(volPos, YY, ZZ, XX * ZZ,
                                                     ZZ, YY * ZZ, 1);
    // Z pass: L=96, stride=1, lines=X*Y, base = l*Z
    edt_pass_kernel<<<2 * XX * YY, 128, 0, stream>>>(volPos, ZZ, 1, XX * YY,
                                                     1, ZZ, 0);
    accum_kernel<<<NB_ACC, 256, 0, stream>>>(net, gt, volPos, volNeg, flags,
                                             partials, b, c, g);
  }
  finish_kernel<<<1, 64, 0, stream>>>(partials, out, NGRP * NB_ACC);
}